// CRFLayer_33423435498441
// MI455X (gfx1250) — compile-verified
//
#include <hip/hip_runtime.h>
#include <hip/hip_bf16.h>
#include <math.h>

// ---------------------------------------------------------------------------
// CRF NLL on MI455X (gfx1250).
//   B=4096 rows, L=4096 steps, T=2 tags.  Memory bound (~218MB -> ~10us floor).
//   Kernel 1: one block per row, 128 threads = 128 segments of 32 steps.
//             - async global->LDS staging of emissions+tags (coalesced,
//               ASYNCcnt), padded LDS layout for conflict-free reads.
//             - per-segment 2x2 log-space matrix (associative scan summary),
//             - 7-level intra-block tree combine -> per-row NLL into ws.
//   Kernel 2: single wave reduces 4096 NLLs to the mean with
//             v_wmma_f32_16x16x4_f32 (ones-matrix reduction) + shfl_xor.
// ---------------------------------------------------------------------------

#define CRF_L   4096
#define CRF_C   32      // steps per segment
#define CRF_S   128     // segments per row  (CRF_S * CRF_C == CRF_L)
#define NEGINF  (-1.0e30f)

// padded LDS strides (bytes) for bank-conflict-free per-segment reads
#define E_SEG_STRIDE 520   // 32 steps * 8B + 264B pad
#define T_SEG_STRIDE 136   // 32 tags  * 4B + 8B pad

// ---------------- CDNA5 async global->LDS copy (8 bytes / lane) ------------
typedef int v2i_as __attribute__((vector_size(8)));   // builtin's element type

__device__ __forceinline__ void async_copy8(void* dst_lds, const void* src_glb) {
#if __has_builtin(__builtin_amdgcn_global_load_async_to_lds_b64)
  __builtin_amdgcn_global_load_async_to_lds_b64(
      (__attribute__((address_space(1))) v2i_as*)const_cast<void*>(src_glb),
      (__attribute__((address_space(3))) v2i_as*)dst_lds,
      0, 0);
#else
  unsigned loff = (unsigned)(unsigned long long)
      (__attribute__((address_space(3))) char*)dst_lds;
  asm volatile("global_load_async_to_lds_b64 %0, %1, off"
               :: "v"(loff), "v"(src_glb) : "memory");
#endif
}

__device__ __forceinline__ void wait_async_all() {
#if __has_builtin(__builtin_amdgcn_s_wait_asynccnt)
  __builtin_amdgcn_s_wait_asynccnt(0);
#else
  asm volatile("s_wait_asynccnt 0" ::: "memory");
#endif
}

// log(exp(a)+exp(b)), safe for NEGINF sentinels (all finite, no NaNs)
__device__ __forceinline__ float lse2(float a, float b) {
  float mx = fmaxf(a, b);
  float mn = fminf(a, b);
  return mx + __logf(1.0f + __expf(mn - mx));
}

// ---------------------------------------------------------------------------
// Kernel 1: one block (128 threads) per batch row.
// ---------------------------------------------------------------------------
__global__ __launch_bounds__(CRF_S) void crf_row_kernel(
    const float* __restrict__ emissions,       // [B][L][2] f32
    const int* __restrict__ tags,              // [B][L]    i32
    const unsigned char* __restrict__ mask,    // [B][L]    bool (0/1 bytes)
    const float* __restrict__ trans,           // [2][2]
    const float* __restrict__ startT,          // [2]
    const float* __restrict__ endT,            // [2]
    float* __restrict__ nll_out)               // [B]
{
  __shared__ unsigned long long e_lds[(CRF_S * E_SEG_STRIDE) / 8]; // 66560 B
  __shared__ unsigned long long t_lds[(CRF_S * T_SEG_STRIDE) / 8]; // 17408 B
  __shared__ float sG[CRF_S][4];
  __shared__ float sGold[CRF_S];
  __shared__ int   sCnt[CRF_S];
  __shared__ int   sLast[CRF_S];
  __shared__ int   sHas[CRF_S];

  const int t = threadIdx.x;           // segment id, 0..127
  const int b = blockIdx.x;            // row id

  const char* erow = (const char*)emissions + (size_t)b * CRF_L * 8;
  const char* trow = (const char*)tags      + (size_t)b * CRF_L * 4;
  const unsigned char* mrow = mask + (size_t)b * CRF_L;

  // ---- stage emissions: 32 async b64 per thread, coalesced per wave ----
  #pragma unroll 4
  for (int i = 0; i < 32; ++i) {
    int g   = i * CRF_S + t;                 // global step
    int seg = g >> 5, c = g & 31;
    async_copy8((char*)e_lds + seg * E_SEG_STRIDE + c * 8,
                erow + (size_t)g * 8);
  }
  // ---- stage tags (pairs of i32): 16 async b64 per thread ----
  #pragma unroll 4
  for (int i = 0; i < 16; ++i) {
    int p   = i * CRF_S + t;                 // pair index
    int g   = p << 1;
    int seg = g >> 5, c = g & 31;
    async_copy8((char*)t_lds + seg * T_SEG_STRIDE + c * 4,
                trow + (size_t)g * 4);
  }

  // ---- overlap: direct loads of mask window + parameters ----
  unsigned mu[8];
  {
    const unsigned* mw = (const unsigned*)(mrow + (size_t)t * CRF_C);
    #pragma unroll
    for (int j = 0; j < 8; ++j) mu[j] = mw[j];
  }
  const float T00 = trans[0], T01 = trans[1], T10 = trans[2], T11 = trans[3];

  wait_async_all();
  __syncthreads();

  // ---- per-segment scan ----
  const char* eseg = (const char*)e_lds + t * E_SEG_STRIDE;
  const char* tseg = (const char*)t_lds + t * T_SEG_STRIDE;

  float G00 = 0.f, G01 = NEGINF, G10 = NEGINF, G11 = 0.f;  // identity
  float gsum = 0.f;
  int cnt = 0, lastTag = 0, has = 0;
  int prevTag = 0;
  if (t > 0) {  // last tag of previous segment (raw adjacency for gold trans)
    prevTag = *(const int*)((const char*)t_lds + (t - 1) * T_SEG_STRIDE + 31 * 4);
  }
  const int g0 = t * CRF_C;

  #pragma unroll 4
  for (int c = 0; c < CRF_C; ++c) {
    float2 e = *(const float2*)(eseg + c * 8);
    int   tg = *(const int*)(tseg + c * 4);
    int    m = (mu[c >> 2] >> ((c & 3) * 8)) & 1;
    int    g = g0 + c;

    if (m) { ++cnt; lastTag = tg; has = 1; }
    if (g > 0 && m) {
      // gold partial: emit[g][tg] + trans[tg][prevTag]
      float emitT = tg ? e.y : e.x;
      float tr    = tg ? (prevTag ? T11 : T10) : (prevTag ? T01 : T00);
      gsum += emitT + tr;
      // forward: G <- A_g (x) G, A_g[i][j] = T[i][j] + e[i]
      float n00 = lse2(T00 + e.x + G00, T01 + e.x + G10);
      float n01 = lse2(T00 + e.x + G01, T01 + e.x + G11);
      float n10 = lse2(T10 + e.y + G00, T11 + e.y + G10);
      float n11 = lse2(T10 + e.y + G01, T11 + e.y + G11);
      G00 = n00; G01 = n01; G10 = n10; G11 = n11;
    }
    prevTag = tg;
  }

  sG[t][0] = G00; sG[t][1] = G01; sG[t][2] = G10; sG[t][3] = G11;
  sGold[t] = gsum; sCnt[t] = cnt; sLast[t] = lastTag; sHas[t] = has;
  __syncthreads();

  // ---- tree combine over segments (left = earlier, right = later) ----
  for (int st = 1; st < CRF_S; st <<= 1) {
    int idx = t * (st << 1);
    if (idx + st < CRF_S) {
      float L00 = sG[idx][0], L01 = sG[idx][1], L10 = sG[idx][2], L11 = sG[idx][3];
      float R00 = sG[idx+st][0], R01 = sG[idx+st][1], R10 = sG[idx+st][2], R11 = sG[idx+st][3];
      float n00 = lse2(R00 + L00, R01 + L10);
      float n01 = lse2(R00 + L01, R01 + L11);
      float n10 = lse2(R10 + L00, R11 + L10);
      float n11 = lse2(R10 + L01, R11 + L11);
      sG[idx][0] = n00; sG[idx][1] = n01; sG[idx][2] = n10; sG[idx][3] = n11;
      sGold[idx] += sGold[idx + st];
      sCnt[idx]  += sCnt[idx + st];
      if (sHas[idx + st]) { sLast[idx] = sLast[idx + st]; sHas[idx] = 1; }
    }
    __syncthreads();
  }

  // ---- finalize (thread 0) ----
  if (t == 0) {
    float2 e0 = *(const float2*)((const char*)e_lds + 0);
    int   tg0 = *(const int*)((const char*)t_lds + 0);
    float st0 = startT[0], st1 = startT[1];
    float en0 = endT[0],   en1 = endT[1];

    float a0 = st0 + e0.x, a1 = st1 + e0.y;                // alpha_0
    float f0 = lse2(sG[0][0] + a0, sG[0][1] + a1);         // G_total (x) alpha_0
    float f1 = lse2(sG[0][2] + a0, sG[0][3] + a1);
    float fwd = lse2(f0 + en0, f1 + en1);

    float gold = (tg0 ? st1 + e0.y : st0 + e0.x)
               + sGold[0]
               + (sLast[0] ? en1 : en0);
    float len = fmaxf((float)sCnt[0], 1.0f);
    nll_out[b] = (fwd - gold) / len;
  }
}

// ---------------------------------------------------------------------------
// Kernel 2: single-wave mean reduction via V_WMMA_F32_16X16X4_F32.
//   Lane partials p[0..31] placed in A (K=0 / K=2 slots per the ISA layout),
//   B = ones  =>  D[m][n] = p[m] + p[m+16];  sum rows + shfl_xor(16).
// ---------------------------------------------------------------------------
typedef float v2f __attribute__((ext_vector_type(2)));
typedef float v8f __attribute__((ext_vector_type(8)));

__global__ __launch_bounds__(32) void crf_mean_kernel(
    const float* __restrict__ nll, float* __restrict__ out, int n)
{
  int lane = threadIdx.x;                 // exactly one wave32, EXEC all ones
  float p = 0.f;
  for (int i = lane; i < n; i += 32) p += nll[i];   // deterministic order

  v2f a;  a.x = p;   a.y = 0.f;
  v2f bv; bv.x = 1.f; bv.y = 1.f;
  v8f c = {};
  c = __builtin_amdgcn_wmma_f32_16x16x4_f32(
        /*neg_a=*/false, a, /*neg_b=*/false, bv,
        /*c_mod=*/(short)0, c, /*reuse_a=*/false, /*reuse_b=*/false);

  float s = c[0] + c[1] + c[2] + c[3] + c[4] + c[5] + c[6] + c[7];
  float tot = s + __shfl_xor(s, 16, 32);  // lanes 0..15 <-> 16..31
  if (lane == 0) out[0] = tot / (float)n;
}

// ---------------------------------------------------------------------------
extern "C" void kernel_launch(void* const* d_in, const int* in_sizes, int n_in,
                              void* d_out, int out_size, void* d_ws, size_t ws_size,
                              hipStream_t stream) {
  const float*         emissions = (const float*)d_in[0];
  const int*           tags      = (const int*)d_in[1];
  const unsigned char* maskp     = (const unsigned char*)d_in[2];
  const float*         trans     = (const float*)d_in[3];
  const float*         startT    = (const float*)d_in[4];
  const float*         endT      = (const float*)d_in[5];

  const int B = in_sizes[1] / CRF_L;     // mask/tags are B*L
  float* nll = (float*)d_ws;             // B floats of scratch

  crf_row_kernel<<<B, CRF_S, 0, stream>>>(emissions, tags, maskp,
                                          trans, startT, endT, nll);
  crf_mean_kernel<<<1, 32, 0, stream>>>(nll, (float*)d_out, B);
}